// memory_73332271612496
// MI455X (gfx1250) — compile-verified
//
#include <hip/hip_runtime.h>
#include <hip/hip_bf16.h>
#include <math.h>

// Problem constants (match reference)
constexpr int MSZ = 131072;   // memory slots
constexpr int DSZ = 256;      // key dim
constexpr int BSZ = 256;      // batch
constexpr int KSZ = 256;      // choose_k
constexpr float ALPHA = 0.5f;
constexpr float BETA  = 1e-8f;

typedef float v2f __attribute__((ext_vector_type(2)));
typedef float v8f __attribute__((ext_vector_type(8)));

// ---------------------------------------------------------------------------
// helpers
// ---------------------------------------------------------------------------
__device__ __forceinline__ unsigned ordkey(float f) {
    unsigned u = __float_as_uint(f);
    return (u & 0x80000000u) ? ~u : (u | 0x80000000u);
}

__device__ __forceinline__ float block_sum_256(float v, float* red, int tid) {
    red[tid] = v;
    __syncthreads();
    for (int s = 128; s > 0; s >>= 1) {
        if (tid < s) red[tid] += red[tid + s];
        __syncthreads();
    }
    float r = red[0];
    __syncthreads();
    return r;
}

// descending bitonic sort of 256 (key,payload) pairs in LDS
__device__ void bitonic_desc_256(unsigned* su, int* si, int tid) {
    for (int ks = 2; ks <= 256; ks <<= 1) {
        for (int j = ks >> 1; j > 0; j >>= 1) {
            int ixj = tid ^ j;
            if (ixj > tid) {
                unsigned a = su[tid], c = su[ixj];
                bool down = ((tid & ks) == 0);
                bool sw = down ? (a < c) : (a > c);
                if (sw) {
                    su[tid] = c; su[ixj] = a;
                    int t = si[tid]; si[tid] = si[ixj]; si[ixj] = t;
                }
            }
            __syncthreads();
        }
    }
}

// ---------------------------------------------------------------------------
// Kernel 0a: mean of memory_hist.  Kernel 0b: precompute log(hist+BETA)
// ---------------------------------------------------------------------------
__global__ void mean_hist_kernel(const float* __restrict__ hist, float* __restrict__ meanh) {
    __shared__ float red[256];
    int tid = threadIdx.x;
    float s = 0.f;
    for (int m = tid; m < MSZ; m += 256) s += hist[m];
    float tot = block_sum_256(s, red, tid);
    if (tid == 0) *meanh = tot / (float)MSZ;
}

__global__ void prep_loghist(const float* __restrict__ hist, float* __restrict__ lh) {
    for (int i = blockIdx.x * blockDim.x + threadIdx.x; i < MSZ;
         i += gridDim.x * blockDim.x)
        lh[i] = logf(hist[i] + BETA);
}

// ---------------------------------------------------------------------------
// Kernel 1: sims[b,m] = dot(q[b,:], mk[m,:])  via V_WMMA_F32_16X16X4_F32
// One wave owns one mTile and 8 bTiles (8 accumulators = 64 VGPRs, under the
// 128-VGPR/wave budget => no spill).  Each mk fragment feeds 8 WMMAs, so mk
// (128 MB) is streamed from HBM only twice.  The a-fragment loads are batched
// in a separate phase from the WMMAs so they clause together and the WMMAs
// drain against staggered s_wait_loadcnt values instead of serializing.
// ---------------------------------------------------------------------------
__global__ void gemm_sims_wmma(const float* __restrict__ q,
                               const float* __restrict__ mk,
                               float* __restrict__ sims) {
    int wave  = (blockIdx.x * blockDim.x + threadIdx.x) >> 5;
    int lane  = threadIdx.x & 31;
    int mTile = wave >> 1;        // 8192 m-tiles
    int bg    = wave & 1;         // b-tile group: tiles [bg*8, bg*8+8)
    int r16   = lane & 15;
    int khalf = lane >> 4;        // lanes 16-31 hold K=2,3 of each 4-step

    const float* mrow  = mk + (size_t)(mTile * 16 + r16) * DSZ + 2 * khalf;
    const float* qbase = q  + (size_t)(bg * 128 + r16) * DSZ + 2 * khalf;

    v8f acc[8] = {};
    for (int kk = 0; kk < DSZ; kk += 4) {
        v2f bfrag = *(const v2f*)(mrow + kk);
        v2f af[8];
        #pragma unroll
        for (int t = 0; t < 8; ++t)
            af[t] = *(const v2f*)(qbase + t * 16 * DSZ + kk);
        #pragma unroll
        for (int t = 0; t < 8; ++t)
            acc[t] = __builtin_amdgcn_wmma_f32_16x16x4_f32(
                false, af[t], false, bfrag, (short)0, acc[t], false, false);
    }

    int mcol = mTile * 16 + r16;
    #pragma unroll
    for (int t = 0; t < 8; ++t) {
        int browBase = (bg * 8 + t) * 16 + 8 * khalf;
        #pragma unroll
        for (int r = 0; r < 8; ++r) {
            sims[(size_t)(browBase + r) * MSZ + mcol] = acc[t][r];
        }
    }
}

// ---------------------------------------------------------------------------
// Kernel 2: fused dual-variant radix-select top-K (descending) per row.
// Variant 0: key = sims + log(hist+BETA); variant 1: additionally -2*is_wrong.
// Both variants share the sims row reads (histograms side by side in LDS),
// then each selection is bitonic-sorted descending.
// ---------------------------------------------------------------------------
__global__ void select_topk2(const float* __restrict__ sims,
                             const float* __restrict__ lh,
                             const float* __restrict__ values,
                             const float* __restrict__ label,
                             int* __restrict__ idx0,
                             int* __restrict__ idx1) {
    int b = blockIdx.x;
    int tid = threadIdx.x;

    __shared__ unsigned h0[256], h1[256];
    __shared__ int s_bucket0, s_rem0, s_bucket1, s_rem1;
    __shared__ unsigned cntG0, cntT0, cntG1, cntT1;
    __shared__ unsigned sel_u0[256], sel_u1[256];
    __shared__ int      sel_i0[256], sel_i1[256];

    const float lab = label[b];
    const float* srow = sims + (size_t)b * MSZ;

    unsigned prefix0 = 0, pmask0 = 0, prefix1 = 0, pmask1 = 0;
    int rem0 = KSZ, rem1 = KSZ;

    for (int pass = 0; pass < 4; ++pass) {
        int shift = 24 - 8 * pass;
        h0[tid] = 0; h1[tid] = 0;
        __syncthreads();
        for (int m = tid; m < MSZ; m += 256) {
            float base = srow[m] + lh[m];
            float isw  = fminf(fabsf(lab - values[m]), 1.0f);
            unsigned u0 = ordkey(base);
            unsigned u1 = ordkey(base - 2.0f * isw);
            if ((u0 & pmask0) == prefix0) atomicAdd(&h0[(u0 >> shift) & 255u], 1u);
            if ((u1 & pmask1) == prefix1) atomicAdd(&h1[(u1 >> shift) & 255u], 1u);
        }
        __syncthreads();
        if (tid == 0) {
            unsigned cum = 0; int d = 255;
            for (; d >= 0; --d) {
                unsigned c = h0[d];
                if (cum + c >= (unsigned)rem0) break;
                cum += c;
            }
            if (d < 0) d = 0;
            s_bucket0 = d; s_rem0 = rem0 - (int)cum;
        } else if (tid == 32) {
            unsigned cum = 0; int d = 255;
            for (; d >= 0; --d) {
                unsigned c = h1[d];
                if (cum + c >= (unsigned)rem1) break;
                cum += c;
            }
            if (d < 0) d = 0;
            s_bucket1 = d; s_rem1 = rem1 - (int)cum;
        }
        __syncthreads();
        prefix0 |= ((unsigned)s_bucket0) << shift;  pmask0 |= 0xFFu << shift;
        prefix1 |= ((unsigned)s_bucket1) << shift;  pmask1 |= 0xFFu << shift;
        rem0 = s_rem0;  rem1 = s_rem1;
        __syncthreads();
    }

    if (tid == 0) { cntG0 = 0; cntT0 = 0; cntG1 = 0; cntT1 = 0; }
    sel_u0[tid] = 0; sel_i0[tid] = 0;
    sel_u1[tid] = 0; sel_i1[tid] = 0;
    __syncthreads();
    int G0 = KSZ - rem0, G1 = KSZ - rem1;
    for (int m = tid; m < MSZ; m += 256) {
        float base = srow[m] + lh[m];
        float isw  = fminf(fabsf(lab - values[m]), 1.0f);
        unsigned u0 = ordkey(base);
        unsigned u1 = ordkey(base - 2.0f * isw);
        if (u0 > prefix0) {
            unsigned p = atomicAdd(&cntG0, 1u);
            if (p < (unsigned)G0) { sel_u0[p] = u0; sel_i0[p] = m; }
        } else if (u0 == prefix0) {
            unsigned p = atomicAdd(&cntT0, 1u);
            if (p < (unsigned)rem0) { sel_u0[G0 + p] = u0; sel_i0[G0 + p] = m; }
        }
        if (u1 > prefix1) {
            unsigned p = atomicAdd(&cntG1, 1u);
            if (p < (unsigned)G1) { sel_u1[p] = u1; sel_i1[p] = m; }
        } else if (u1 == prefix1) {
            unsigned p = atomicAdd(&cntT1, 1u);
            if (p < (unsigned)rem1) { sel_u1[G1 + p] = u1; sel_i1[G1 + p] = m; }
        }
    }
    __syncthreads();

    bitonic_desc_256(sel_u0, sel_i0, tid);
    bitonic_desc_256(sel_u1, sel_i1, tid);

    idx0[b * KSZ + tid] = sel_i0[tid];
    idx1[b * KSZ + tid] = sel_i1[tid];
}

// ---------------------------------------------------------------------------
// Kernel 2b: select B oldest slots by age (single block radix select)
// ---------------------------------------------------------------------------
__global__ void select_oldest(const float* __restrict__ age, int* __restrict__ oldest) {
    int tid = threadIdx.x;
    __shared__ unsigned hc[256];
    __shared__ int s_bucket, s_rem;
    __shared__ unsigned cntG, cntT;
    __shared__ unsigned sel_u[256];
    __shared__ int      sel_i[256];

    unsigned prefix = 0, pmask = 0;
    int rem = BSZ;
    for (int pass = 0; pass < 4; ++pass) {
        int shift = 24 - 8 * pass;
        hc[tid] = 0;
        __syncthreads();
        for (int m = tid; m < MSZ; m += 256) {
            unsigned u = ordkey(age[m]);
            if ((u & pmask) == prefix) atomicAdd(&hc[(u >> shift) & 255u], 1u);
        }
        __syncthreads();
        if (tid == 0) {
            unsigned cum = 0; int d = 255;
            for (; d >= 0; --d) {
                unsigned c = hc[d];
                if (cum + c >= (unsigned)rem) break;
                cum += c;
            }
            if (d < 0) d = 0;
            s_bucket = d; s_rem = rem - (int)cum;
        }
        __syncthreads();
        prefix |= ((unsigned)s_bucket) << shift;
        pmask  |= 0xFFu << shift;
        rem = s_rem;
        __syncthreads();
    }

    if (tid == 0) { cntG = 0; cntT = 0; }
    sel_u[tid] = 0; sel_i[tid] = 0;
    __syncthreads();
    int G = BSZ - rem;
    for (int m = tid; m < MSZ; m += 256) {
        unsigned u = ordkey(age[m]);
        if (u > prefix) {
            unsigned p = atomicAdd(&cntG, 1u);
            if (p < (unsigned)G) { sel_u[p] = u; sel_i[p] = m; }
        } else if (u == prefix) {
            unsigned p = atomicAdd(&cntT, 1u);
            if (p < (unsigned)rem) { sel_u[G + p] = u; sel_i[G + p] = m; }
        }
    }
    __syncthreads();
    bitonic_desc_256(sel_u, sel_i, tid);
    oldest[tid] = sel_i[tid];
}

// ---------------------------------------------------------------------------
// Kernel 3: per-row posterior result, reset flag, and EM-state initialization
// ---------------------------------------------------------------------------
__global__ void gather_init(const float* __restrict__ q,
                            const float* __restrict__ label,
                            const float* __restrict__ mk,
                            const float* __restrict__ values,
                            const float* __restrict__ hist,
                            const float* __restrict__ sims,
                            const int* __restrict__ idx0,
                            const int* __restrict__ idx1,
                            float* __restrict__ khat,
                            float* __restrict__ hhat,
                            float* __restrict__ vhat,
                            int* __restrict__ reset,
                            float* __restrict__ result) {
    int b = blockIdx.x;
    int tid = threadIdx.x;
    __shared__ float red[256];
    __shared__ int idxsh[256];

    // result[b] = sum(post0 * red_vals) over top-k (no label hint)
    int i0 = idx0[b * KSZ + tid];
    float s = sims[(size_t)b * MSZ + i0];
    float j = expf(s - 1.0f) * (hist[i0] * ALPHA + BETA);
    float v = values[i0];
    float S  = block_sum_256(j, red, tid);
    float JV = block_sum_256(j * v, red, tid);
    if (tid == 0) {
        result[b] = JV / S;
        // reset if top-1 retrieved value disagrees with label
        int top = idx0[b * KSZ];
        float hint = 1.0f - fminf(fabsf(label[b] - values[top]), 1.0f);
        reset[b] = (hint == 0.0f) ? 1 : 0;
    }

    // label-conditioned selection -> EM state
    int i1 = idx1[b * KSZ + tid];
    hhat[b * KSZ + tid] = ALPHA * hist[i1];
    vhat[b * KSZ + tid] = values[i1];
    idxsh[tid] = i1;
    __syncthreads();
    // coalesced gather of K key rows: thread = d, loop over k
    for (int k = 0; k < KSZ; ++k) {
        khat[((size_t)b * KSZ + k) * DSZ + tid] = mk[(size_t)idxsh[k] * DSZ + tid];
    }
}

// ---------------------------------------------------------------------------
// Kernel 4: exclusive scan of reset flags (rank of each reset row)
// ---------------------------------------------------------------------------
__global__ void prefix_reset(const int* __restrict__ reset, int* __restrict__ tpre) {
    if (threadIdx.x == 0) {
        int c = 0;
        for (int b = 0; b < BSZ; ++b) { tpre[b] = c; c += reset[b]; }
    }
}

// ---------------------------------------------------------------------------
// Kernel 5: 3-step EM key update, one block per batch row
// ---------------------------------------------------------------------------
__global__ void em_kernel(const float* __restrict__ q,
                          float* __restrict__ khat,
                          float* __restrict__ hhat) {
    int b = blockIdx.x;
    int tid = threadIdx.x;
    __shared__ float qsh[256];
    __shared__ float red[256];
    __shared__ float inv[256];

    float* kh = khat + (size_t)b * KSZ * DSZ;
    qsh[tid] = q[(size_t)b * DSZ + tid];
    float hh = hhat[b * KSZ + tid];
    float gm = 0.0f;
    __syncthreads();

    for (int step = 0; step < 3; ++step) {
        float* row = kh + (size_t)tid * DSZ;
        float s = 0.f;
        for (int d = 0; d < DSZ; ++d) s += qsh[d] * row[d];
        float joint = expf(s - 1.0f) * (hh + BETA);
        float S = block_sum_256(joint, red, tid);
        float ng = joint / S;
        float diff = ng - gm;
        float DS = block_sum_256(diff, red, tid);
        hh += DS;                              // broadcast add over K
        float r = diff / hh;
        for (int d = 0; d < DSZ; ++d) row[d] = row[d] * (1.0f - r) + r * qsh[d];
        __syncthreads();
        // L1 normalize over K (dim=1 of [B,K,D]) : per-(b,d) column
        float cs = 0.f;
        for (int k = 0; k < KSZ; ++k) cs += fabsf(kh[(size_t)k * DSZ + tid]);
        inv[tid] = 1.0f / fmaxf(cs, 1e-12f);
        __syncthreads();
        for (int d = 0; d < DSZ; ++d) row[d] *= inv[d];
        gm = ng;
        __syncthreads();
    }
    hhat[b * KSZ + tid] = hh;
}

// ---------------------------------------------------------------------------
// Kernel 6: bulk copies into d_out
// ---------------------------------------------------------------------------
__global__ void copy_key(const float* __restrict__ mk, float* __restrict__ outk) {
    size_t n = (size_t)MSZ * DSZ;
    for (size_t i = (size_t)blockIdx.x * blockDim.x + threadIdx.x; i < n;
         i += (size_t)gridDim.x * blockDim.x)
        outk[i] = mk[i];
}

__global__ void copy_rest(const float* __restrict__ values,
                          const float* __restrict__ age,
                          const float* __restrict__ hist,
                          float* __restrict__ ov, float* __restrict__ oa,
                          float* __restrict__ oh) {
    for (int i = blockIdx.x * blockDim.x + threadIdx.x; i < MSZ;
         i += gridDim.x * blockDim.x) {
        ov[i] = values[i];
        oa[i] = age[i] + 1.0f;
        oh[i] = hist[i];
    }
}

// ---------------------------------------------------------------------------
// Kernel 7: scatter updates (faithful masked_scatter semantics)
// ---------------------------------------------------------------------------
__global__ void scatter_out(const float* __restrict__ q,
                            const float* __restrict__ label,
                            const int* __restrict__ idx1,
                            const int* __restrict__ oldest,
                            const int* __restrict__ reset,
                            const int* __restrict__ tpre,
                            const float* __restrict__ khat,
                            const float* __restrict__ hhat,
                            const float* __restrict__ vhat,
                            const float* __restrict__ meanh,
                            float* __restrict__ outk, float* __restrict__ ov,
                            float* __restrict__ oa, float* __restrict__ oh) {
    int b = blockIdx.x;
    int k = threadIdx.x;
    int dst;
    const float* src;
    float val, h;
    if (reset[b]) {
        int t = tpre[b];                 // rank among reset rows (src consumed seq.)
        dst = oldest[t];
        src = q + (size_t)t * DSZ;
        val = label[t];
        h   = hhat[t * KSZ + k];
    } else {
        dst = idx1[b * KSZ + k];
        src = khat + ((size_t)b * KSZ + k) * DSZ;
        val = vhat[b * KSZ + k];
        h   = *meanh;                    // non-reset rows get mean(hist)
    }
    oa[dst] = 0.0f;
    ov[dst] = val;
    oh[dst] = h;
    for (int d = 0; d < DSZ; ++d) outk[(size_t)dst * DSZ + d] = src[d];
}

// ---------------------------------------------------------------------------
// host launcher
// ---------------------------------------------------------------------------
extern "C" void kernel_launch(void* const* d_in, const int* in_sizes, int n_in,
                              void* d_out, int out_size, void* d_ws, size_t ws_size,
                              hipStream_t stream) {
    const float* q      = (const float*)d_in[0];
    const float* label  = (const float*)d_in[1];
    const float* mk     = (const float*)d_in[2];
    const float* values = (const float*)d_in[3];
    const float* age    = (const float*)d_in[4];
    const float* hist   = (const float*)d_in[5];

    // carve workspace
    float* w = (float*)d_ws;
    size_t o = 0;
    float* sims  = w + o; o += (size_t)BSZ * MSZ;        // 128 MB
    float* khat  = w + o; o += (size_t)BSZ * KSZ * DSZ;  // 64 MB
    float* lh    = w + o; o += MSZ;                      // log(hist+BETA)
    float* hhat  = w + o; o += (size_t)BSZ * KSZ;
    float* vhat  = w + o; o += (size_t)BSZ * KSZ;
    int*   idx0  = (int*)(w + o); o += (size_t)BSZ * KSZ;
    int*   idx1  = (int*)(w + o); o += (size_t)BSZ * KSZ;
    int*   oldest= (int*)(w + o); o += BSZ;
    int*   reset = (int*)(w + o); o += BSZ;
    int*   tpre  = (int*)(w + o); o += BSZ;
    float* meanh = w + o; o += 1;

    // output layout: result | new_key | new_values | new_age | new_hist
    float* out    = (float*)d_out;
    float* o_res  = out;
    float* o_key  = o_res + BSZ;
    float* o_val  = o_key + (size_t)MSZ * DSZ;
    float* o_age  = o_val + MSZ;
    float* o_hist = o_age + MSZ;

    mean_hist_kernel<<<1, 256, 0, stream>>>(hist, meanh);
    prep_loghist<<<256, 256, 0, stream>>>(hist, lh);

    // 8192 mTiles x 2 bGroups = 16384 waves, 8 waves per 256-thread block
    gemm_sims_wmma<<<16384 / 8, 256, 0, stream>>>(q, mk, sims);

    select_topk2<<<BSZ, 256, 0, stream>>>(sims, lh, values, label, idx0, idx1);
    select_oldest<<<1, 256, 0, stream>>>(age, oldest);

    gather_init<<<BSZ, 256, 0, stream>>>(q, label, mk, values, hist, sims,
                                         idx0, idx1, khat, hhat, vhat, reset, o_res);
    prefix_reset<<<1, 32, 0, stream>>>(reset, tpre);
    em_kernel<<<BSZ, 256, 0, stream>>>(q, khat, hhat);

    copy_key<<<8192, 256, 0, stream>>>(mk, o_key);
    copy_rest<<<512, 256, 0, stream>>>(values, age, hist, o_val, o_age, o_hist);
    scatter_out<<<BSZ, 256, 0, stream>>>(q, label, idx1, oldest, reset, tpre,
                                         khat, hhat, vhat, meanh,
                                         o_key, o_val, o_age, o_hist);
}